// ConvCaps_81432579932497
// MI455X (gfx1250) — compile-verified
//
#include <hip/hip_runtime.h>
#include <hip/hip_bf16.h>
#include <math.h>

// ConvCaps (EM-routing capsule conv) fused kernel for gfx1250 (MI455X).
// One workgroup per output position (256 WGs), 8 wave32s per WG.
// All routing state lives in LDS (62.5 KB / WG, 320 KB per WGP available).
// Matrix work uses V_WMMA_F32_16X16X4_F32 (f32 precision required: EM routing
// takes log/rcp of variances near EPS; K=4 matches the pose contraction and
// chains as an accumulator for the K=288 / K=32 routing GEMMs).

typedef __attribute__((ext_vector_type(2))) float v2f;
typedef __attribute__((ext_vector_type(8))) float v8f;

#define BK      288      // K*K*B_CAPS
#define CC      32       // C_CAPS
#define EPSF    1e-8f
#define LAMBDAF 1e-3f
#define LN2PI_F 1.8378770664093453f

__device__ __forceinline__ v8f wmma4(v2f a, v2f b, v8f c) {
  // D(16x16,f32) = A(16x4,f32) x B(4x16,f32) + C
  return __builtin_amdgcn_wmma_f32_16x16x4_f32(
      /*neg_a=*/false, a, /*neg_b=*/false, b,
      /*c_mod=*/(short)0, c, /*reuse_a=*/false, /*reuse_b=*/false);
}

__global__ __launch_bounds__(256) void convcaps_em_kernel(
    const float* __restrict__ x,       // (4,16,16,544)
    const float* __restrict__ w,       // (288,4,4)
    const float* __restrict__ beta_u,  // (32,)
    const float* __restrict__ beta_a,  // (32,)
    float* __restrict__ out)           // (4,8,8,544)
{
  __shared__ float s_v[BK * 16];       // votes v[i][m]            18432 B
  __shared__ float s_r[BK * CC];       // routing r[i][c]          36864 B
  __shared__ float s_a[BK];            // a_in[i]                   1152 B
  __shared__ float s_mu[CC * 16];      // mu[c][m]                  2048 B
  __shared__ float s_ev2[CC * 16];     // E[v^2][c][m]              2048 B
  __shared__ float s_sig[CC * 16];     // sigma^2[c][m]             2048 B
  __shared__ float s_rsum[CC];         //                             128 B
  __shared__ float s_aout[CC];         //                             128 B
  __shared__ float s_gamma[CC];        //                             128 B
  __shared__ float s_part[8 * CC];     // reduction scratch          1024 B

  const int tid  = threadIdx.x;
  const int wv   = tid >> 5;   // wave id, 0..7 (wave32)
  const int lane = tid & 31;
  const int n  = blockIdx.x;           // position
  const int b  = n >> 6;
  const int oy = (n >> 3) & 7;
  const int ox = n & 7;

  // ---- Phase A: gather activations a_in[i] (zero for padded taps) ----
  for (int i = tid; i < BK; i += 256) {
    const int tap = i >> 5, bcap = i & 31;
    const int iy = oy * 2 + (tap / 3) - 1;
    const int ix = ox * 2 + (tap % 3) - 1;
    float a = 0.0f;
    if ((unsigned)iy < 16u && (unsigned)ix < 16u)
      a = x[(((b * 16 + iy) * 16) + ix) * 544 + 512 + bcap];
    s_a[i] = a;
  }

  // ---- Phase B: votes v[i] = p_in[i] @ W[i] via f32 WMMA, 4 capsules per
  //      instruction packed block-diagonally ----
  {
    const int rl = lane & 15;             // A-row / B-col within tile
    const int kk = (lane >> 4) << 1;      // K lane offset: {0,2}
    // Only lanes 0-7 and 24-31 hold diagonal-block outputs:
    //   row>>2 (= reg{0-3:lo,4-7:hi} + 8*(lane>>4)) must equal col>>2.
    const bool part = (((lane >> 3) ^ (lane >> 4)) & 1) == 0;
    const bool hiRegs = (lane >> 2) & 1;                 // use acc[4..7]
    const int  capInGrp = ((lane >> 4) << 1) | ((lane >> 2) & 1);
    for (int g = wv; g < BK / 4; g += 8) {           // 72 groups, 9 per wave
      const int tap = (g * 4) >> 5;                  // uniform within group
      const int iy = oy * 2 + (tap / 3) - 1;
      const int ix = ox * 2 + (tap % 3) - 1;
      const bool inb = ((unsigned)iy < 16u) && ((unsigned)ix < 16u);
      // A[4a+p, q] = p_in[4g+a, p, q]
      v2f av; av.x = 0.0f; av.y = 0.0f;
      const int icap = g * 4 + (rl >> 2);
      const int bcap = icap & 31;
      if (inb) {
        const float* px =
            x + (((b * 16 + iy) * 16) + ix) * 544 + bcap * 16 + (rl & 3) * 4 + kk;
        av.x = px[0]; av.y = px[1];
      }
      // B[q, 4a+r] = W[4g+a, q, r]
      const float* pw = w + icap * 16 + (rl & 3);
      v2f bv; bv.x = pw[kk * 4]; bv.y = pw[(kk + 1) * 4];
      v8f acc = {};
      acc = wmma4(av, bv, acc);
      // harvest diagonal 4x4 blocks: D[4a+p, 4a+r] -> v[4g+a, p*4+r]
      const int base = (g * 4 + capInGrp) * 16 + (lane & 3);
      if (part && !hiRegs) {
        s_v[base +  0] = acc[0];
        s_v[base +  4] = acc[1];
        s_v[base +  8] = acc[2];
        s_v[base + 12] = acc[3];
      }
      if (part && hiRegs) {
        s_v[base +  0] = acc[4];
        s_v[base +  4] = acc[5];
        s_v[base +  8] = acc[6];
        s_v[base + 12] = acc[7];
      }
    }
  }

  // ---- init r = 1/C ----
  for (int t = tid; t < BK * CC; t += 256) s_r[t] = 1.0f / 32.0f;
  __syncthreads();

  // ---- EM routing: 3 iterations ----
  for (int it = 0; it < 3; ++it) {
    // (1) r'' = (r * a_in) / (sum_c + eps), in place
    for (int i = tid; i < BK; i += 256) {
      const float a = s_a[i];
      float sum = 0.0f;
      for (int c = 0; c < CC; ++c) {
        const float vv = s_r[i * CC + c] * a;
        s_r[i * CC + c] = vv;
        sum += vv;
      }
      const float inv = 1.0f / (sum + EPSF);
      for (int c = 0; c < CC; ++c) s_r[i * CC + c] *= inv;
    }
    // zero GEMM accumulators
    for (int t = tid; t < CC * 16; t += 256) { s_mu[t] = 0.0f; s_ev2[t] = 0.0f; }
    __syncthreads();

    // (2) r_sum[c] = sum_i r''[i,c]
    {
      const int c = tid & 31, seg = tid >> 5;
      float p = 0.0f;
      for (int i = seg * 36; i < seg * 36 + 36; ++i) p += s_r[i * CC + c];
      s_part[seg * 32 + c] = p;
    }
    __syncthreads();
    if (tid < CC) {
      float s = 0.0f;
      for (int k = 0; k < 8; ++k) s += s_part[k * 32 + tid];
      s_rsum[tid] = s;
    }

    // (3) split-K WMMA GEMMs: mu_raw[c,m] = sum_i r''[i,c] v[i,m],
    //     ev2_raw[c,m] = sum_i r''[i,c] v[i,m]^2   (K = 288 chained)
    {
      const int which = wv & 3;       // 0,1: mu ctile0/1 ; 2,3: ev2 ctile0/1
      const int ctile = which & 1;
      const int sq    = which >> 1;
      const int khalf = wv >> 2;
      const int rl = lane & 15;
      const int kk = (lane >> 4) << 1;
      v8f acc = {};
#pragma unroll 2
      for (int kc = 0; kc < 36; ++kc) {
        const int kb = (khalf * 36 + kc) * 4;
        v2f av, bv;
        av.x = s_r[(kb + kk) * CC + ctile * 16 + rl];       // A = r''^T
        av.y = s_r[(kb + kk + 1) * CC + ctile * 16 + rl];
        float b0 = s_v[(kb + kk) * 16 + rl];                 // B = v or v^2
        float b1 = s_v[(kb + kk + 1) * 16 + rl];
        if (sq) { b0 *= b0; b1 *= b1; }
        bv.x = b0; bv.y = b1;
        acc = wmma4(av, bv, acc);
      }
      float* dst = sq ? s_ev2 : s_mu;
#pragma unroll
      for (int j = 0; j < 8; ++j) {
        const int r0 = ctile * 16 + j + ((lane >> 4) << 3);
        atomicAdd(&dst[r0 * 16 + (lane & 15)], acc[j]);      // ds_add_f32
      }
    }
    __syncthreads();

    // (4) finalize mu, sigma^2 (S = sum_i coeff = rsum/(rsum+eps))
    for (int t = tid; t < CC * 16; t += 256) {
      const int c = t >> 4;
      const float s = s_rsum[c];
      const float inv = 1.0f / (s + EPSF);
      const float S = s * inv;
      const float mu = s_mu[t] * inv;
      const float ev2 = s_ev2[t] * inv;
      s_mu[t] = mu;
      s_sig[t] = ev2 - (2.0f - S) * mu * mu + EPSF;
    }
    __syncthreads();

    // (5) a_out[c] and e-step bias gamma[c]
    if (tid < CC) {
      const int c = tid;
      float slog = 0.0f, g2 = 0.0f;
      for (int m = 0; m < 16; ++m) {
        const float sg = s_sig[c * 16 + m];
        const float lg = 0.5f * __logf(sg);
        slog += lg;
        const float mu = s_mu[c * 16 + m];
        g2 += -(mu * mu) / (2.0f * sg) - lg;
      }
      const float cost = (16.0f * beta_u[c] + slog) * s_rsum[c];
      const float z = LAMBDAF * (beta_a[c] - cost);
      const float ao = 1.0f / (1.0f + __expf(-z));
      s_aout[c] = ao;
      s_gamma[c] = g2 - 8.0f * LN2PI_F + __logf(ao);
    }
    __syncthreads();

    if (it < 2) {
      // (6) e-step as GEMM (K=32): ln_ap[i,c] =
      //     sum_m v^2[i,m]*(-1/(2s[c,m])) + v[i,m]*(mu[c,m]/s[c,m]) + gamma[c]
      {
        const int rl = lane & 15;
        const int kk = (lane >> 4) << 1;
        for (int tp = wv; tp < 36; tp += 8) {         // 18 i-tiles x 2 c-tiles
          const int itile = tp >> 1, ctile = tp & 1;
          const int ig = itile * 16 + rl;
          const int cg = ctile * 16 + rl;
          v8f acc = {};
          for (int kc = 0; kc < 8; ++kc) {
            const int k0 = kc * 4 + kk;               // never straddles 16
            v2f av, bv;
            if (k0 < 16) {
              const float v0 = s_v[ig * 16 + k0], v1 = s_v[ig * 16 + k0 + 1];
              av.x = v0 * v0; av.y = v1 * v1;
              bv.x = -0.5f / s_sig[cg * 16 + k0];
              bv.y = -0.5f / s_sig[cg * 16 + k0 + 1];
            } else {
              const int m0 = k0 - 16;
              av.x = s_v[ig * 16 + m0]; av.y = s_v[ig * 16 + m0 + 1];
              bv.x = s_mu[cg * 16 + m0] / s_sig[cg * 16 + m0];
              bv.y = s_mu[cg * 16 + m0 + 1] / s_sig[cg * 16 + m0 + 1];
            }
            acc = wmma4(av, bv, acc);
          }
#pragma unroll
          for (int j = 0; j < 8; ++j) {
            const int i = itile * 16 + j + ((lane >> 4) << 3);
            const int c = ctile * 16 + (lane & 15);
            s_r[i * CC + c] = acc[j] + s_gamma[c];
          }
        }
      }
      __syncthreads();
      // (7) softmax over c
      for (int i = tid; i < BK; i += 256) {
        float mx = -3.0e38f;
        for (int c = 0; c < CC; ++c) mx = fmaxf(mx, s_r[i * CC + c]);
        float sum = 0.0f;
        for (int c = 0; c < CC; ++c) {
          const float e = __expf(s_r[i * CC + c] - mx);
          s_r[i * CC + c] = e;
          sum += e;
        }
        const float inv = 1.0f / sum;
        for (int c = 0; c < CC; ++c) s_r[i * CC + c] *= inv;
      }
      __syncthreads();
    }
  }

  // ---- output: poses (c*16+m) then activations ----
  for (int t = tid; t < 544; t += 256) {
    out[n * 544 + t] = (t < 512) ? s_mu[t] : s_aout[t - 512];
  }
}

extern "C" void kernel_launch(void* const* d_in, const int* in_sizes, int n_in,
                              void* d_out, int out_size, void* d_ws, size_t ws_size,
                              hipStream_t stream) {
  (void)in_sizes; (void)n_in; (void)out_size; (void)d_ws; (void)ws_size;
  const float* x  = (const float*)d_in[0];
  const float* w  = (const float*)d_in[1];
  const float* bu = (const float*)d_in[2];
  const float* ba = (const float*)d_in[3];
  float* out = (float*)d_out;
  convcaps_em_kernel<<<dim3(256), dim3(256), 0, stream>>>(x, w, bu, ba, out);
}